// NormalizedHistogram_31250182045884
// MI455X (gfx1250) — compile-verified
//
#include <hip/hip_runtime.h>

// ---------------------------------------------------------------------------
// NormalizedHistogram for MI455X (gfx1250, wave32)
//   input : [32, 512, 512, 3] float32 in [0,1]
//   output: [32, 256, 3] float32, per-(image,channel) histogram / (512*512)
// Memory-bound scatter-reduce: float4 (b128) loads -> LDS-privatized
// ds_add_u32 histograms -> single global_atomic_add_u32 flush -> finalize.
// ---------------------------------------------------------------------------

#define NH_NBINS   256
#define NH_C       3
#define NH_BATCH   32
#define NH_HW      (512 * 512)                    // 262144 pixels / image
#define NH_FLOATS  (NH_HW * NH_C)                 // 786432 floats / image
#define NH_VEC4    (NH_FLOATS / 4)                // 196608 float4 / image
#define NH_THREADS 256                            // 8 wave32 per block
#define NH_V4PT    16                             // float4 per thread
#define NH_BLKS    (NH_VEC4 / (NH_THREADS * NH_V4PT)) // 48 blocks / image
#define NH_CNT_PB  (NH_C * NH_NBINS)              // 768 counters / image
#define NH_CNT_TOT (NH_BATCH * NH_CNT_PB)         // 24576 counters total
#define NH_PFDIST  8                              // prefetch distance (iters)

__global__ void nh_zero_counters(unsigned int* __restrict__ g) {
    int i = blockIdx.x * blockDim.x + threadIdx.x;
    if (i < NH_CNT_TOT) g[i] = 0u;
}

__global__ __launch_bounds__(NH_THREADS)
void nh_hist_kernel(const float* __restrict__ in,
                    unsigned int* __restrict__ gcount) {
    __shared__ unsigned int lhist[NH_CNT_PB];

    const int tid = threadIdx.x;

    // zero the block-private histogram (768 counters, 3 per thread)
    #pragma unroll
    for (int k = tid; k < NH_CNT_PB; k += NH_THREADS) lhist[k] = 0u;
    __syncthreads();

    const int b = blockIdx.y;                               // image index
    const float4* __restrict__ src =
        (const float4*)(in + (size_t)b * NH_FLOATS);
    const int base = blockIdx.x * (NH_THREADS * NH_V4PT);   // vec4 base in image

    #pragma unroll
    for (int it = 0; it < NH_V4PT; ++it) {
        const int i = base + it * NH_THREADS + tid;         // float4 index

        // gfx1250 prefetch path (global_prefetch_b8), kept in-bounds:
        if (it < NH_V4PT - NH_PFDIST) {
            __builtin_prefetch((const void*)(src + i + NH_PFDIST * NH_THREADS),
                               0, 1);
        }

        const float4 v = src[i];                            // global_load_b128

        // channel of flat float index 4*i + j is (i + j) % 3  (4 == 1 mod 3)
        int c0 = i % 3;
        int c1 = c0 + 1; if (c1 == 3) c1 = 0;
        int c2 = c1 + 1; if (c2 == 3) c2 = 0;

        // bin = clip(floor(v * 256), 0, 255); trunc==floor for v >= 0,
        // and identical after clamping for v < 0.
        int b0 = (int)(v.x * 256.0f); b0 = b0 < 0 ? 0 : (b0 > 255 ? 255 : b0);
        int b1 = (int)(v.y * 256.0f); b1 = b1 < 0 ? 0 : (b1 > 255 ? 255 : b1);
        int b2 = (int)(v.z * 256.0f); b2 = b2 < 0 ? 0 : (b2 > 255 ? 255 : b2);
        int b3 = (int)(v.w * 256.0f); b3 = b3 < 0 ? 0 : (b3 > 255 ? 255 : b3);

        // ds_add_u32 (non-returning LDS atomics)
        atomicAdd(&lhist[c0 * NH_NBINS + b0], 1u);
        atomicAdd(&lhist[c1 * NH_NBINS + b1], 1u);
        atomicAdd(&lhist[c2 * NH_NBINS + b2], 1u);
        atomicAdd(&lhist[c0 * NH_NBINS + b3], 1u);
    }
    __syncthreads();

    // flush private histogram -> global integer counters for this image
    // (global_atomic_add_u32, non-returning -> STOREcnt path)
    unsigned int* __restrict__ gb = gcount + (size_t)b * NH_CNT_PB;
    #pragma unroll
    for (int k = tid; k < NH_CNT_PB; k += NH_THREADS) {
        const unsigned int cnt = lhist[k];
        atomicAdd(&gb[k], cnt);
    }
}

__global__ void nh_finalize(const unsigned int* __restrict__ gcount,
                            float* __restrict__ out) {
    int i = blockIdx.x * blockDim.x + threadIdx.x;
    if (i >= NH_CNT_TOT) return;
    const int b   = i / NH_CNT_PB;
    const int r   = i - b * NH_CNT_PB;
    const int c   = r / NH_NBINS;
    const int bin = r - c * NH_NBINS;
    // output layout [B, NBINS, C]; normalize by H*W
    out[((size_t)b * NH_NBINS + bin) * NH_C + c] =
        (float)gcount[i] * (1.0f / (float)NH_HW);
}

extern "C" void kernel_launch(void* const* d_in, const int* in_sizes, int n_in,
                              void* d_out, int out_size, void* d_ws, size_t ws_size,
                              hipStream_t stream) {
    const float*  in     = (const float*)d_in[0];
    float*        out    = (float*)d_out;
    unsigned int* gcount = (unsigned int*)d_ws;   // 24576 u32 = 96 KB scratch

    (void)in_sizes; (void)n_in; (void)out_size; (void)ws_size;

    // 1) zero integer counters (d_ws is not re-initialized between replays)
    nh_zero_counters<<<NH_CNT_TOT / NH_THREADS, NH_THREADS, 0, stream>>>(gcount);

    // 2) main histogram: grid = (48 blocks/image, 32 images), 256 thr = 8 wave32
    dim3 grid(NH_BLKS, NH_BATCH, 1);
    nh_hist_kernel<<<grid, NH_THREADS, 0, stream>>>(in, gcount);

    // 3) normalize + transpose to [B, NBINS, C]
    nh_finalize<<<NH_CNT_TOT / NH_THREADS, NH_THREADS, 0, stream>>>(gcount, out);
}